// Solution_38869454029188
// MI455X (gfx1250) — compile-verified
//
#include <hip/hip_runtime.h>

#define B_N    2048
#define S_SP   100
#define NR_R   800
#define OUTW   905          // 3 + 800 + 1 + 1 + 100
#define RGAS   8314.46261815324f
#define ATM    101325.0f

typedef __attribute__((ext_vector_type(2))) float v2f;
typedef __attribute__((ext_vector_type(8))) float v8f;

__device__ __forceinline__ float blockReduceSum(float v, float* red) {
    const int tid = threadIdx.x;
    red[tid] = v;
    __syncthreads();
    #pragma unroll
    for (int off = 64; off > 0; off >>= 1) {
        if (tid < off) red[tid] += red[tid + off];
        __syncthreads();
    }
    float r = red[0];
    __syncthreads();
    return r;
}

// ---------------------------------------------------------------------------
// Kernel 1: one 128-thread block per batch row. Computes Y-normalization,
// mean W, rho, X, C (-> workspace for WMMA GEMM), NASA-poly thermo sums,
// viscosity mixing (PHI), thermal conductivity, and mixture diffusion.
// ---------------------------------------------------------------------------
__global__ __launch_bounds__(128) void props_kernel(
    const float* __restrict__ TPY,   const float* __restrict__ mw,
    const float* __restrict__ nlow,  const float* __restrict__ nhigh,
    const float* __restrict__ viscP, const float* __restrict__ condP,
    const float* __restrict__ bdifP, float* __restrict__ out,
    float* __restrict__ Cmat)
{
    __shared__ float sY[S_SP], sX[S_SP], sXe[S_SP], sMw[S_SP], sVi[S_SP], sCo[S_SP];
    __shared__ float sRed[128];

    const int b   = blockIdx.x;
    const int tid = threadIdx.x;
    const float* row = TPY + (size_t)b * (S_SP + 2);

    const float T    = fmaxf(row[0], 200.0f);
    const float P    = row[1];
    const float logT = logf(T);

    // ---- load Y (clamped), mw -------------------------------------------
    for (int s = tid; s < S_SP; s += 128) {
        sY[s]  = fmaxf(row[2 + s], 0.0f);
        sMw[s] = mw[s];
    }
    __syncthreads();

    float ys = 0.0f;
    for (int s = tid; s < S_SP; s += 128) ys += sY[s];
    ys = blockReduceSum(ys, sRed);
    const float inv_ys = 1.0f / ys;

    float yw = 0.0f;
    for (int s = tid; s < S_SP; s += 128) {
        float y = sY[s] * inv_ys;
        sY[s] = y;
        yw += y / sMw[s];
    }
    yw = blockReduceSum(yw, sRed);

    const float meanW = 1.0f / yw;
    const float rho   = P / (RGAS * T) * meanW;

    for (int s = tid; s < S_SP; s += 128) {
        float invw = 1.0f / sMw[s];
        float x = sY[s] * meanW * invw;
        sX[s]  = x;
        sXe[s] = fmaxf(x, 1e-12f);
        Cmat[(size_t)b * S_SP + s] = sY[s] * rho * invw;   // A-matrix for WMMA
    }
    __syncthreads();

    // ---- NASA polynomial thermo -----------------------------------------
    const float* nasa = (T <= 1000.0f) ? nlow : nhigh;
    const float T2 = T * T, T3 = T2 * T, T4 = T2 * T2;
    const float logPatm = logf(P / ATM);

    float cpm = 0.0f, hm = 0.0f, sm = 0.0f;
    for (int s = tid; s < S_SP; s += 128) {
        const float* a = nasa + s * 7;
        float cp = a[0] + a[1]*T + a[2]*T2 + a[3]*T3 + a[4]*T4;
        float H  = (a[0] + 0.5f*a[1]*T + (1.0f/3.0f)*a[2]*T2 + 0.25f*a[3]*T3
                    + 0.2f*a[4]*T4 + a[5]/T) * (RGAS * T);
        float S0 = (a[0]*logT + a[1]*T + 0.5f*a[2]*T2 + (1.0f/3.0f)*a[3]*T3
                    + 0.25f*a[4]*T4 + a[6]) * RGAS;
        float Ssp = S0 - RGAS * (logf(sX[s]) + logPatm);
        float invw = 1.0f / sMw[s];
        cpm += cp  * sX[s];
        hm  += H   * sY[s] * invw;
        sm  += Ssp * sY[s] * invw;
    }
    cpm = blockReduceSum(cpm, sRed) * RGAS;
    hm  = blockReduceSum(hm,  sRed);
    sm  = blockReduceSum(sm,  sRed);
    if (tid == 0) {
        float* o = out + (size_t)b * OUTW;
        o[0] = cpm; o[1] = hm; o[2] = sm;
    }

    // ---- transport polynomials in log(T) (descending powers) ------------
    const float lt = logT, lt2 = lt*lt;
    float tr[5] = { lt2*lt2, lt2*lt, lt2, lt, 1.0f };
    for (int s = tid; s < S_SP; s += 128) {
        float v = 0.0f, c = 0.0f;
        #pragma unroll
        for (int p = 0; p < 5; ++p) {
            v = fmaf(tr[p], viscP[p * S_SP + s], v);
            c = fmaf(tr[p], condP[p * S_SP + s], c);
        }
        sVi[s] = v; sCo[s] = c;
    }
    __syncthreads();

    // ---- viscosity mixing (PHI), conductivity, sum(Xe*W) -----------------
    float viscAcc = 0.0f, xc = 0.0f, xoc = 0.0f, xw = 0.0f;
    for (int i = tid; i < S_SP; i += 128) {
        const float xi = sX[i], vi = sVi[i], mwi = sMw[i];
        float denom = 0.0f;
        for (int j = 0; j < S_SP; ++j) {
            float rw  = mwi / sMw[j];                       // W_i / W_j
            float er  = vi / sVi[j];                        // eta_i / eta_j
            float t   = 1.0f + sqrtf(er) * sqrtf(sqrtf(1.0f / rw)); // (Wj/Wi)^0.25
            float phi = 0.35355339059327373f * rsqrtf(1.0f + rw) * t * t;
            denom = fmaf(phi, sX[j], denom);
        }
        viscAcc += xi * vi / denom;
        xc  += xi * sCo[i];
        xoc += xi / sCo[i];
        xw  += sXe[i] * mwi;
    }
    viscAcc = blockReduceSum(viscAcc, sRed);
    xc      = blockReduceSum(xc,  sRed);
    xoc     = blockReduceSum(xoc, sRed);
    xw      = blockReduceSum(xw,  sRed);
    if (tid == 0) {
        float* o = out + (size_t)b * OUTW;
        o[803] = viscAcc;
        o[804] = 0.5f * (xc + 1.0f / xoc);
    }

    // ---- mixture diffusion: XjWj / XjDjk / meanW / P * ATM ---------------
    const float scale = (1.0f / meanW) / P * ATM;
    for (int k = tid; k < S_SP; k += 128) {
        float sumInv = 0.0f, diag = 0.0f;
        for (int j = 0; j < S_SP; ++j) {
            float d = 0.0f;
            #pragma unroll
            for (int p = 0; p < 5; ++p)
                d = fmaf(tr[p], bdifP[p * (S_SP * S_SP) + j * S_SP + k], d);
            float inv = sXe[j] / d;
            sumInv += inv;
            if (j == k) diag = inv;
        }
        float XjDjk = sumInv - diag;
        float XjWj  = xw - sXe[k] * sMw[k];
        out[(size_t)b * OUTW + 805 + k] = XjWj / XjDjk * scale;
    }
}

// ---------------------------------------------------------------------------
// Kernel 2: C_M2 = (C @ eff) * i3b + (1 - i3b) via V_WMMA_F32_16X16X4_F32.
// grid = (50 N-tiles, 32), block = 128 (4 waves); wave w does M-tile
// blockIdx.y*4+w. K=100 accumulated as 25 WMMA k-steps of 4.
// ---------------------------------------------------------------------------
__global__ __launch_bounds__(128) void gemm_cm_kernel(
    const float* __restrict__ Cmat, const float* __restrict__ eff,
    const float* __restrict__ i3b,  float* __restrict__ out)
{
    const int tn   = blockIdx.x;                       // N tile (0..49)
    const int tm   = blockIdx.y * 4 + (threadIdx.x >> 5); // M tile (0..127)
    const int lane = threadIdx.x & 31;
    const int half = lane >> 4;                        // 0: K lo pair, 1: K hi pair
    const int l16  = lane & 15;

    const int rowA = tm * 16 + l16;                    // A row held by this lane
    const int col  = tn * 16 + l16;                    // B/D column held by this lane

    v8f acc = {};
    #pragma unroll 5
    for (int k0 = 0; k0 < S_SP; k0 += 4) {
        const int ka = k0 + 2 * half;
        v2f a, bb;
        a.x  = Cmat[rowA * S_SP + ka];
        a.y  = Cmat[rowA * S_SP + ka + 1];
        bb.x = eff[ka * NR_R + col];
        bb.y = eff[(ka + 1) * NR_R + col];
        acc = __builtin_amdgcn_wmma_f32_16x16x4_f32(
            /*neg_a=*/false, a, /*neg_b=*/false, bb,
            /*c_mod=*/(short)0, acc, /*reuse_a=*/false, /*reuse_b=*/false);
    }

    const float f = i3b[col];
    const float g = 1.0f - f;
    #pragma unroll
    for (int v = 0; v < 8; ++v) {
        int m = tm * 16 + v + 8 * half;                // D layout: M = v + 8*half
        out[(size_t)m * OUTW + 3 + col] = acc[v] * f + g;
    }
}

extern "C" void kernel_launch(void* const* d_in, const int* in_sizes, int n_in,
                              void* d_out, int out_size, void* d_ws, size_t ws_size,
                              hipStream_t stream) {
    const float* TPY   = (const float*)d_in[0];
    const float* mw    = (const float*)d_in[1];
    const float* nlow  = (const float*)d_in[2];
    const float* nhigh = (const float*)d_in[3];
    const float* eff   = (const float*)d_in[4];
    const float* i3b   = (const float*)d_in[5];
    const float* viscP = (const float*)d_in[6];
    const float* condP = (const float*)d_in[7];
    const float* bdifP = (const float*)d_in[8];
    float* out  = (float*)d_out;
    float* Cmat = (float*)d_ws;                        // 2048*100 floats = 800 KB

    props_kernel<<<B_N, 128, 0, stream>>>(TPY, mw, nlow, nhigh,
                                          viscP, condP, bdifP, out, Cmat);
    gemm_cm_kernel<<<dim3(50, 32), 128, 0, stream>>>(Cmat, eff, i3b, out);
}